// PointLaplacianLoss_8117488189442
// MI455X (gfx1250) — compile-verified
//
#include <hip/hip_runtime.h>

// ---------------------------------------------------------------------------
// PointLaplacianLoss for MI455X (gfx1250, wave32).
//
// Roofline: inputs are ~192KB/cloud -> everything is L2-resident (192MB L2);
// the problem is compute-bound on B*N^2 = 134M distance evals + top-k tests.
// We feed the distance math to the matrix pipe: with A pre-scaled by -2 and
// C preloaded with |q|^2 + |c|^2, V_WMMA_F32_16X16X4_F32 (K padded 3->4)
// produces the finished squared-distance tile, so the VALU only runs the
// top-10 selection filter (1 compare per candidate on the common path).
//
// Round-2 refinements: row scan reads the 64B LDS row via 4x ds_load_b128
// instead of 16x ds_load_b32, and the redundant trailing dscnt wait is gone
// (DS is in-order within a wave; may-alias keeps compiler ordering).
// ---------------------------------------------------------------------------

typedef __attribute__((ext_vector_type(2))) float v2f;
typedef __attribute__((ext_vector_type(8))) float v8f;

#define NB    2
#define NPTS  8192
#define KNN   10
#define WAVES 8          // 256-thread block = 8 wave32
#define QPW   32         // queries per wave (two 16x16 WMMA tiles)
#define QPB   (WAVES * QPW)   // 256 queries per block

// Prep: per-point squared norms of point1 into workspace; zero the output
// accumulator (must happen every call - harness poisons d_out once).
__global__ void prep_kernel(const float* __restrict__ p1,
                            float* __restrict__ sq,
                            float* __restrict__ out) {
    int i = blockIdx.x * blockDim.x + threadIdx.x;
    if (i == 0) out[0] = 0.0f;
    if (i < NB * NPTS) {
        float x = p1[i * 3 + 0], y = p1[i * 3 + 1], z = p1[i * 3 + 2];
        sq[i] = x * x + y * y + z * z;
    }
}

__global__ __launch_bounds__(QPB) void knn_lap_kernel(
    const float* __restrict__ p1, const float* __restrict__ p2,
    const float* __restrict__ sq, float* __restrict__ out) {
    // Per-wave private 32x16 f32 distance tile (8 waves -> 16KB of 320KB LDS).
    __shared__ float tile[WAVES][QPW][16];

    const int tid  = threadIdx.x;
    const int w    = tid >> 5;
    const int lane = tid & 31;
    const int hi   = lane >> 4;   // 0: lanes 0-15, 1: lanes 16-31
    const int lo   = lane & 15;

    const int b     = blockIdx.x >> 5;                    // 32 blocks / batch
    const int qbase = (blockIdx.x & 31) * QPB + w * QPW;  // wave's 32 queries

    const float* pb1 = p1 + (size_t)b * NPTS * 3;
    const float* pb2 = p2 + (size_t)b * NPTS * 3;
    const float* sqb = sq + (size_t)b * NPTS;

    // A fragments (16x4 f32, 2 VGPRs): lanes 0-15 carry K0,K1 = -2x,-2y;
    // lanes 16-31 carry K2,K3 = -2z, 0 (K padded to 4).
    v2f a0, a1;
    {
        int q0 = qbase + lo;
        int q1 = qbase + 16 + lo;
        float x0 = pb1[q0 * 3 + 0], y0 = pb1[q0 * 3 + 1], z0 = pb1[q0 * 3 + 2];
        float x1 = pb1[q1 * 3 + 0], y1 = pb1[q1 * 3 + 1], z1 = pb1[q1 * 3 + 2];
        a0[0] = hi ? (-2.0f * z0) : (-2.0f * x0);
        a0[1] = hi ? 0.0f         : (-2.0f * y0);
        a1[0] = hi ? (-2.0f * z1) : (-2.0f * x1);
        a1[1] = hi ? 0.0f         : (-2.0f * y1);
    }

    // |q|^2 per C/D row: VGPR j holds row j + 8*hi.
    float sqq0[8], sqq1[8];
#pragma unroll
    for (int j = 0; j < 8; ++j) {
        sqq0[j] = sqb[qbase + 8 * hi + j];
        sqq1[j] = sqb[qbase + 16 + 8 * hi + j];
    }

    // Sorted top-10 (ascending) register list per lane; lane owns one query.
    float nd[KNN];
    int   ni[KNN];
#pragma unroll
    for (int i = 0; i < KNN; ++i) { nd[i] = 3.0e38f; ni[i] = 0; }
    const int myq = qbase + lane;

    for (int ct = 0; ct < NPTS / 16; ++ct) {
        // B fragment (4x16 f32): candidate coords, mirrored A layout.
        const int cand = ct * 16 + lo;
        const float cx = pb1[cand * 3 + 0];
        const float cy = pb1[cand * 3 + 1];
        const float cz = pb1[cand * 3 + 2];
        const float cs = sqb[cand];
        v2f bf;
        bf[0] = hi ? cz   : cx;
        bf[1] = hi ? 0.0f : cy;

        // C = |q|^2 + |c|^2 so D is the finished squared distance.
        v8f c0, c1;
#pragma unroll
        for (int j = 0; j < 8; ++j) { c0[j] = sqq0[j] + cs; c1[j] = sqq1[j] + cs; }

        v8f d0 = __builtin_amdgcn_wmma_f32_16x16x4_f32(
            false, a0, false, bf, (short)0, c0, false, false);
        v8f d1 = __builtin_amdgcn_wmma_f32_16x16x4_f32(
            false, a1, false, bf, (short)0, c1, false, false);

        // Scatter D tiles to this wave's LDS slab (row = query-in-wave).
#pragma unroll
        for (int j = 0; j < 8; ++j) {
            tile[w][j + 8 * hi][lo]      = d0[j];
            tile[w][16 + j + 8 * hi][lo] = d1[j];
        }
        // DS ops are in-order within a wave: a dscnt wait is sufficient,
        // no workgroup barrier needed (tile slab is wave-private).
        asm volatile("s_wait_dscnt 0" ::: "memory");

        // Pull the whole 64B row with 4x ds_load_b128, then run the
        // insertion filter entirely out of registers.
        const float4* row4 = (const float4*)&tile[w][lane][0];
        float dv[16];
#pragma unroll
        for (int q = 0; q < 4; ++q) {
            float4 r = row4[q];
            dv[q * 4 + 0] = r.x;
            dv[q * 4 + 1] = r.y;
            dv[q * 4 + 2] = r.z;
            dv[q * 4 + 3] = r.w;
        }
#pragma unroll
        for (int n = 0; n < 16; ++n) {
            float d  = dv[n];
            int   ci = ct * 16 + n;
            if (ci != myq && d < nd[KNN - 1]) {
                nd[KNN - 1] = d;
                ni[KNN - 1] = ci;
                // One bubble pass restores sorted order (list was sorted).
#pragma unroll
                for (int s = KNN - 1; s > 0; --s) {
                    if (nd[s] < nd[s - 1]) {
                        float td = nd[s]; nd[s] = nd[s - 1]; nd[s - 1] = td;
                        int   ti = ni[s]; ni[s] = ni[s - 1]; ni[s - 1] = ti;
                    }
                }
            }
        }
    }

    // Laplacian difference for this lane's query (neighbor data is L2-hot).
    float s1x = 0.f, s1y = 0.f, s1z = 0.f;
    float s2x = 0.f, s2y = 0.f, s2z = 0.f;
#pragma unroll
    for (int i = 0; i < KNN; ++i) {
        int j = ni[i];
        s1x += pb1[j * 3 + 0]; s1y += pb1[j * 3 + 1]; s1z += pb1[j * 3 + 2];
        s2x += pb2[j * 3 + 0]; s2y += pb2[j * 3 + 1]; s2z += pb2[j * 3 + 2];
    }
    const float inv_k = 1.0f / (float)KNN;
    // lap1 - lap2 = (mean1 - mean2) - (p1 - p2)
    float dx = (s1x - s2x) * inv_k - (pb1[myq * 3 + 0] - pb2[myq * 3 + 0]);
    float dy = (s1y - s2y) * inv_k - (pb1[myq * 3 + 1] - pb2[myq * 3 + 1]);
    float dz = (s1z - s2z) * inv_k - (pb1[myq * 3 + 2] - pb2[myq * 3 + 2]);
    float part = fabsf(dx) + fabsf(dy) + fabsf(dz);

    // wave32 reduction, then one atomic per wave.
#pragma unroll
    for (int off = 16; off > 0; off >>= 1)
        part += __shfl_xor(part, off, 32);
    if (lane == 0)
        atomicAdd(out, part * (1.0f / (float)(NB * NPTS * 3)));
}

extern "C" void kernel_launch(void* const* d_in, const int* in_sizes, int n_in,
                              void* d_out, int out_size, void* d_ws, size_t ws_size,
                              hipStream_t stream) {
    const float* p1 = (const float*)d_in[0];
    const float* p2 = (const float*)d_in[1];
    float* out = (float*)d_out;
    float* sq  = (float*)d_ws;   // NB*NPTS floats = 64KB of scratch

    prep_kernel<<<(NB * NPTS + 255) / 256, 256, 0, stream>>>(p1, sq, out);
    knn_lap_kernel<<<(NB * NPTS) / QPB, QPB, 0, stream>>>(p1, p2, sq, out);
}